// TreeGuidedAttention_66666482368945
// MI455X (gfx1250) — compile-verified
//
#include <hip/hip_runtime.h>
#include <hip/hip_bf16.h>
#include <math.h>

// ---------------------------------------------------------------------------
// TreeGuidedAttention for MI455X (gfx1250, wave32)
//   B=64, dim=512, depth=9, N=1023 nodes, 512 leaves.
//   HBM-bound on node_states (134 MB); GEMMs done with V_WMMA_F32_16X16X4_F32.
// ---------------------------------------------------------------------------

typedef float v2f __attribute__((ext_vector_type(2)));
typedef float v8f __attribute__((ext_vector_type(8)));

#define DIM     512
#define BATCH   64
#define NNODES  1023
#define NLEAF   512
#define DEPTH   9
#define CH      32      // feature-dims per tree block (== one wave wide)

__device__ __forceinline__ v8f wmma_f32_16x16x4(v2f a, v2f b, v8f c) {
  // 8 args: (neg_a, A, neg_b, B, c_mod, C, reuse_a, reuse_b)
  return __builtin_amdgcn_wmma_f32_16x16x4_f32(false, a, false, b,
                                               (short)0, c, false, false);
}

// ---------------------------------------------------------------------------
// Kernel 1: target[64][512] = hidden @ W_in^T      (WMMA f32, one wave/tile)
//   tile grid: 4 (M/16) x 32 (N/16) = 128 waves
// ---------------------------------------------------------------------------
__global__ void k_gemm_win(const float* __restrict__ hidden,
                           const float* __restrict__ W_in,
                           float* __restrict__ target) {
  const int lane  = threadIdx.x & 31;
  const int wave  = threadIdx.x >> 5;
  const int tile  = blockIdx.x * (blockDim.x >> 5) + wave;
  const int tm    = tile >> 5;            // 0..3
  const int tn    = tile & 31;            // 0..31
  const int m     = tm * 16 + (lane & 15);
  const int n     = tn * 16 + (lane & 15);
  const int khalf = (lane >> 4) * 2;      // 0 or 2

  const float* arow = hidden + (size_t)m * DIM;
  const float* brow = W_in   + (size_t)n * DIM;   // B[k][n] = W_in[n][k]

  v8f c = {0.f, 0.f, 0.f, 0.f, 0.f, 0.f, 0.f, 0.f};
  for (int k0 = 0; k0 < DIM; k0 += 4) {
    v2f a = *(const v2f*)(arow + k0 + khalf);
    v2f b = *(const v2f*)(brow + k0 + khalf);
    c = wmma_f32_16x16x4(a, b, c);
  }

  const int row0 = tm * 16 + 8 * (lane >> 4);
  const int col  = tn * 16 + (lane & 15);
#pragma unroll
  for (int r = 0; r < 8; ++r)
    target[(size_t)(row0 + r) * DIM + col] = c[r];
}

// ---------------------------------------------------------------------------
// Kernel 2: scores[b][n] = dot(target[b], node_states[b][n])
//   Streams 134 MB of node_states with coalesced b128 loads.
//   One wave per node row; target[b] staged once in LDS per block.
// ---------------------------------------------------------------------------
__global__ void k_scores(const float* __restrict__ target,
                         const float* __restrict__ node_states,
                         float* __restrict__ scores) {
  __shared__ float tgt[DIM];
  const int b = blockIdx.y;
  for (int i = threadIdx.x; i < DIM; i += blockDim.x)
    tgt[i] = target[(size_t)b * DIM + i];
  __syncthreads();

  const int wave = threadIdx.x >> 5;
  const int lane = threadIdx.x & 31;
  const int n    = blockIdx.x * (blockDim.x >> 5) + wave;
  if (n >= NNODES) return;

  const float* row = node_states + ((size_t)b * NNODES + n) * DIM;
  // speculative prefetch of the next row this WGP will touch
  __builtin_prefetch(row + 8 * DIM, 0, 0);

  float acc = 0.f;
#pragma unroll
  for (int it = 0; it < 4; ++it) {
    const int off = it * 128 + lane * 4;              // 16B/lane, coalesced
    float4 v = *(const float4*)(row + off);           // global_load_b128
    float4 t = *(const float4*)(&tgt[off]);           // ds_load_b128
    acc += v.x * t.x + v.y * t.y + v.z * t.z + v.w * t.w;
  }
#pragma unroll
  for (int s = 16; s > 0; s >>= 1)
    acc += __shfl_xor(acc, s, 32);
  if (lane == 0)
    scores[(size_t)b * NNODES + n] = acc;
}

// ---------------------------------------------------------------------------
// Kernel 3: tree recursion, fully LDS-resident.
//   Per-dim independence: block = (batch b, 32-dim chunk). Ping-pong
//   buffers 512xCH / 256xCH in dynamic LDS (~100 KB of the 320 KB WGP LDS).
// ---------------------------------------------------------------------------
__global__ void k_tree(const float* __restrict__ scores,
                       const float* __restrict__ node_states,
                       float* __restrict__ root) {
  extern __shared__ float smem[];
  float* sc   = smem;                 // 1024 floats (1023 used)
  float* bufA = smem + 1024;          // 512*CH
  float* bufB = bufA + NLEAF * CH;    // 256*CH

  const int b     = blockIdx.y;
  const int jbase = blockIdx.x * CH;

  for (int i = threadIdx.x; i < NNODES; i += blockDim.x)
    sc[i] = scores[(size_t)b * NNODES + i];
  __syncthreads();

  // init: attn[leaf][j] = scores[leaf_start + leaf]  (leaf_start = 511)
  for (int item = threadIdx.x; item < NLEAF * CH; item += blockDim.x)
    bufA[item] = sc[(NLEAF - 1) + (item >> 5)];
  __syncthreads();

  float* cur = bufA;
  float* nxt = bufB;
  for (int d = DEPTH - 1; d >= 0; --d) {
    const int nodes = 1 << d;
    const int base  = nodes - 1;            // first node index at level d
    for (int item = threadIdx.x; item < nodes * CH; item += blockDim.x) {
      const int i = item >> 5;              // node within level
      const int j = item & 31;              // dim within chunk
      const int g = base + i;               // global node index
      const float la = sc[2 * g + 1];
      const float ra = sc[2 * g + 2];
      const float w  = 1.f / (1.f + expf(ra - la));   // sigmoid(la - ra)
      const float lh = cur[(2 * i) * CH + j];
      const float rh = cur[(2 * i + 1) * CH + j];
      const float ns = node_states[((size_t)b * NNODES + g) * DIM + jbase + j];
      nxt[item] = sc[g] * (w * lh + (1.f - w) * rh + ns);
    }
    __syncthreads();
    float* t = cur; cur = nxt; nxt = t;
  }

  if (threadIdx.x < CH)
    root[(size_t)b * DIM + jbase + threadIdx.x] = cur[threadIdx.x];
}

// ---------------------------------------------------------------------------
// Kernel 4: out = tanh(concat([root, hidden]) @ W_out^T)   (WMMA f32)
//   M=64, N=512, K=1024 (first 512 from root, last 512 from hidden)
// ---------------------------------------------------------------------------
__global__ void k_gemm_wout(const float* __restrict__ root,
                            const float* __restrict__ hidden,
                            const float* __restrict__ W_out,
                            float* __restrict__ out) {
  const int lane  = threadIdx.x & 31;
  const int wave  = threadIdx.x >> 5;
  const int tile  = blockIdx.x * (blockDim.x >> 5) + wave;
  const int tm    = tile >> 5;            // 0..3
  const int tn    = tile & 31;            // 0..31
  const int m     = tm * 16 + (lane & 15);
  const int n     = tn * 16 + (lane & 15);
  const int khalf = (lane >> 4) * 2;

  const float* a_lo = root   + (size_t)m * DIM;
  const float* a_hi = hidden + (size_t)m * DIM;
  const float* brow = W_out  + (size_t)n * (2 * DIM);

  v8f c = {0.f, 0.f, 0.f, 0.f, 0.f, 0.f, 0.f, 0.f};
  for (int k0 = 0; k0 < DIM; k0 += 4) {
    v2f a = *(const v2f*)(a_lo + k0 + khalf);
    v2f b = *(const v2f*)(brow + k0 + khalf);
    c = wmma_f32_16x16x4(a, b, c);
  }
  for (int k0 = 0; k0 < DIM; k0 += 4) {
    v2f a = *(const v2f*)(a_hi + k0 + khalf);
    v2f b = *(const v2f*)(brow + DIM + k0 + khalf);
    c = wmma_f32_16x16x4(a, b, c);
  }

  const int row0 = tm * 16 + 8 * (lane >> 4);
  const int col  = tn * 16 + (lane & 15);
#pragma unroll
  for (int r = 0; r < 8; ++r)
    out[(size_t)(row0 + r) * DIM + col] = tanhf(c[r]);
}

// ---------------------------------------------------------------------------
extern "C" void kernel_launch(void* const* d_in, const int* in_sizes, int n_in,
                              void* d_out, int out_size, void* d_ws, size_t ws_size,
                              hipStream_t stream) {
  const float* hidden      = (const float*)d_in[0];   // (64, 512)
  const float* node_states = (const float*)d_in[1];   // (64, 1023, 512)
  const float* W_in        = (const float*)d_in[2];   // (512, 512)
  const float* W_out       = (const float*)d_in[3];   // (512, 1024)
  float*       out         = (float*)d_out;           // (64, 512)

  float* ws     = (float*)d_ws;
  float* target = ws;                                  // 64*512   = 32768 floats
  float* scores = ws + BATCH * DIM;                    // 64*1023 (padded to 64*1024)
  float* root   = ws + BATCH * DIM + BATCH * 1024;     // 64*512
  // total workspace used: 131072 floats = 512 KB

  // 1) target = hidden @ W_in^T            (128 waves, WMMA f32 16x16x4)
  k_gemm_win<<<32, 128, 0, stream>>>(hidden, W_in, target);

  // 2) scores = einsum('bd,bnd->bn')       (streams 134 MB, HBM-bound)
  {
    dim3 grid((NNODES + 7) / 8, BATCH);
    k_scores<<<grid, 256, 0, stream>>>(target, node_states, scores);
  }

  // 3) tree recursion, LDS-resident        (64 batches x 16 dim-chunks)
  {
    const size_t lds = (size_t)(1024 + NLEAF * CH + (NLEAF / 2) * CH) * sizeof(float); // 100 KB
    hipFuncSetAttribute(reinterpret_cast<const void*>(k_tree),
                        hipFuncAttributeMaxDynamicSharedMemorySize, (int)lds);
    dim3 grid(DIM / CH, BATCH);
    k_tree<<<grid, 256, lds, stream>>>(scores, node_states, root);
  }

  // 4) out = tanh(concat([root, hidden]) @ W_out^T)   (128 waves, WMMA f32)
  k_gemm_wout<<<32, 128, 0, stream>>>(root, hidden, W_out, out);
}